// PREF_12549894439511
// MI455X (gfx1250) — compile-verified
//
#include <hip/hip_runtime.h>
#include <math.h>

// ---------------------------------------------------------------------------
// Triplane feature extraction (grid_sample bilinear + cosine-basis contraction)
//
// MI455X strategy: gather-bound workload (no shared-operand GEMM -> WMMA is
// mathematically inapplicable). Win comes from layout: transpose each plane
// from (c,y,x) channel-major (144 cache lines per sampled pixel) to pixel-
// major (y,x, r*16+ch) so each bilinear corner is 576 contiguous bytes.
// All three transposed planes (113 MB) are L2-resident (192 MB).
//
// CDNA5 paths used:
//   - TENSOR_LOAD_TO_LDS (TDM) with hardware LDS padding for the transpose
//     tile load (one DMA descriptor replaces 18 vector loads), S_WAIT_TENSORCNT
//   - GLOBAL_STORE_ASYNC_FROM_LDS_B32 for the transposed write-back (ASYNCcnt)
//   - GLOBAL_PREFETCH_B8 in the gather kernel to hide L2 latency under cosf
// ---------------------------------------------------------------------------

#define CH        16
#define RR        9
#define C144      (CH * RR)          // 144 channels per plane
#define GRID      256
#define NPIX      (GRID * GRID)      // 65536 pixels per plane
#define PIX_STRIDE C144              // floats per pixel in transposed layout

typedef unsigned int u32x4 __attribute__((ext_vector_type(4)));
typedef unsigned int u32x8 __attribute__((ext_vector_type(8)));

// ---------------------- transpose kernel -----------------------------------
// In : P[c0][pix]        with c0 = ch*9 + r        (144 x 65536 f32)
// Out: T[pix][r*16+ch]                              (65536 x 144 f32)
#define TP_PIX     32                // pixels per block (TDM tile_dim0)
#define TP_PAD     33                // LDS row stride (dwords): 32 + 1 pad dword
#define TP_THREADS 256

__global__ __launch_bounds__(TP_THREADS)
void transpose_plane_kernel(const float* __restrict__ P, float* __restrict__ T)
{
    __shared__ float tile[C144 * TP_PAD];     // 19008 B
    const int pixbase = blockIdx.x * TP_PIX;
    const int tid     = threadIdx.x;

    // --- one wave issues a single TDM descriptor for the whole tile --------
    // Tile: dim0 = 32 pixels (contiguous), dim1 = 144 channel rows with
    // stride 65536 elements. data_size = 4B. LDS padding: +1 dword every
    // 32 dwords -> row stride TP_PAD, bank-conflict-free transposed reads.
    const int wave_id = __builtin_amdgcn_readfirstlane((int)(threadIdx.x >> 5));
    if (wave_id == 0) {
        const unsigned long long gaddr =
            (unsigned long long)(const void*)(P + (size_t)pixbase);
        const unsigned lds_base = (unsigned)(size_t)&tile[0];

        u32x4 g0;
        g0.x = 1u;                                        // count=1 (valid, load)
        g0.y = lds_base;                                  // lds_addr (bytes)
        g0.z = (unsigned)(gaddr & 0xffffffffu);           // global_addr[31:0]
        g0.w = (unsigned)((gaddr >> 32) & 0x01ffffffu)    // global_addr[56:32]
             | (2u << 30);                                // type = 2 (image)

        u32x8 g1;
        g1.s0 = (2u << 16)                                // data_size = 4B
              | (1u << 20)                                // pad_enable
              | (4u << 22)                                // pad_interval: 32 dwords
              | (0u << 25);                               // pad_amount: 1 dword
        g1.s1 = 0u;                                       // tensor_dim0[15:0] = 0 (65536)
        g1.s2 = (unsigned)(NPIX >> 16)                    // tensor_dim0[31:16] = 1
              | ((unsigned)C144 << 16);                   // tensor_dim1[15:0] = 144
        g1.s3 = ((unsigned)TP_PIX << 16);                 // tile_dim0 = 32
        g1.s4 = (unsigned)C144;                           // tile_dim1 = 144 (tile_dim2=0)
        g1.s5 = (unsigned)NPIX;                           // tensor_dim0_stride = 65536
        g1.s6 = 0u;
        g1.s7 = 0u;

        asm volatile("tensor_load_to_lds %0, %1"
                     :: "s"(g0), "s"(g1)
                     : "memory");
#if __has_builtin(__builtin_amdgcn_s_wait_tensorcnt)
        __builtin_amdgcn_s_wait_tensorcnt(0);
#else
        asm volatile("s_wait_tensorcnt 0x0" ::: "memory");
#endif
    }
    __syncthreads();

    // --- transposed write-back: LDS -> global directly (async, no VGPR hop).
    // Consecutive tid -> consecutive global addresses (coalesced stores).
    for (int j = tid; j < C144 * TP_PIX; j += TP_THREADS) {
        const int p    = j / C144;
        const int newc = j - p * C144;        // newc = r*16 + ch
        const int r    = newc >> 4;
        const int ch   = newc & 15;
        const int c0   = ch * RR + r;
        const float* gdst = T + (size_t)(pixbase + p) * PIX_STRIDE + newc;
        unsigned lds_off = (unsigned)(size_t)&tile[c0 * TP_PAD + p];
        asm volatile("global_store_async_from_lds_b32 %0, %1, off"
                     :: "v"(gdst), "v"(lds_off)
                     : "memory");
    }
    // s_endpgm performs an implicit wait-idle; ASYNCcnt drains before exit.
}

// ---------------------- per-plane geometry ---------------------------------
struct PlaneGeom {
    int   i00, i01, i10, i11;        // corner pixel indices (y*256 + x)
    float w00, w01, w10, w11;        // bilinear weights
    float basis[RR];                 // cosine basis
};

__device__ __forceinline__ PlaneGeom plane_geom(float gx, float gy, float ci)
{
    PlaneGeom G;
    // align_corners=True: ix = (g+1)/2 * (W-1)
    const float ix  = (gx + 1.f) * 127.5f;
    const float iy  = (gy + 1.f) * 127.5f;
    const float x0f = floorf(ix), y0f = floorf(iy);
    const float wx  = ix - x0f,   wy  = iy - y0f;
    int x0 = min(max((int)x0f, 0), GRID - 1);
    int x1 = min(x0 + 1, GRID - 1);
    int y0 = min(max((int)y0f, 0), GRID - 1);
    int y1 = min(y0 + 1, GRID - 1);
    G.i00 = y0 * GRID + x0;  G.i01 = y0 * GRID + x1;
    G.i10 = y1 * GRID + x0;  G.i11 = y1 * GRID + x1;
    G.w00 = (1.f - wx) * (1.f - wy);  G.w01 = wx * (1.f - wy);
    G.w10 = (1.f - wx) * wy;          G.w11 = wx * wy;
    // c = (ci+1)/2 * (2^(R-1)-1);  basis[r] = cos(pi*(c+0.5)*(2^r-0.5)/2^R)
    const float cc = (ci + 1.f) * 127.5f + 0.5f;
#pragma unroll
    for (int r = 0; r < RR; ++r) {
        const float fr = (float)(1 << r) - 0.5f;
        G.basis[r] = cosf(3.14159265358979323846f * cc * fr * (1.f / 512.f));
    }
    return G;
}

// ---------------------- main kernel (transposed layout) --------------------
__global__ __launch_bounds__(256)
void triplane_kernel(const float* __restrict__ coords,
                     const float* __restrict__ Tu,
                     const float* __restrict__ Tv,
                     const float* __restrict__ Tw,
                     float* __restrict__ out, int N)
{
    const int n = blockIdx.x * blockDim.x + threadIdx.x;
    if (n >= N) return;

    const float cx = coords[3 * n + 0];
    const float cy = coords[3 * n + 1];
    const float cz = coords[3 * n + 2];

    // fu: P(u) sampled at (gx=cy, gy=cz), basis from cx; etc.
    const PlaneGeom Gu = plane_geom(cy, cz, cx);
    const PlaneGeom Gv = plane_geom(cx, cz, cy);
    const PlaneGeom Gw = plane_geom(cx, cy, cz);

    const float*     Ts[3] = { Tu, Tv, Tw };
    const PlaneGeom* Gs[3] = { &Gu, &Gv, &Gw };

    // Prefetch all 12 corner pixels (gfx1250 global_prefetch_b8); their L2
    // latency overlaps the deep cosf dependency chains.
#pragma unroll
    for (int p = 0; p < 3; ++p) {
        const float* T = Ts[p];
        const PlaneGeom& G = *Gs[p];
        __builtin_prefetch(T + (size_t)G.i00 * PIX_STRIDE, 0, 0);
        __builtin_prefetch(T + (size_t)G.i01 * PIX_STRIDE, 0, 0);
        __builtin_prefetch(T + (size_t)G.i10 * PIX_STRIDE, 0, 0);
        __builtin_prefetch(T + (size_t)G.i11 * PIX_STRIDE, 0, 0);
    }

    float acc[CH];
#pragma unroll
    for (int ch = 0; ch < CH; ++ch) acc[ch] = 0.f;

#pragma unroll
    for (int p = 0; p < 3; ++p) {
        const float* T = Ts[p];
        const PlaneGeom& G = *Gs[p];
        const float* b00 = T + (size_t)G.i00 * PIX_STRIDE;
        const float* b01 = T + (size_t)G.i01 * PIX_STRIDE;
        const float* b10 = T + (size_t)G.i10 * PIX_STRIDE;
        const float* b11 = T + (size_t)G.i11 * PIX_STRIDE;
#pragma unroll
        for (int r = 0; r < RR; ++r) {
            const float br  = G.basis[r];
            const float c00 = br * G.w00, c01 = br * G.w01;
            const float c10 = br * G.w10, c11 = br * G.w11;
            // pixel rows are 64B-aligned; each r-group is 16 f32 = 4x float4
            const float4* v00 = (const float4*)(b00 + r * CH);
            const float4* v01 = (const float4*)(b01 + r * CH);
            const float4* v10 = (const float4*)(b10 + r * CH);
            const float4* v11 = (const float4*)(b11 + r * CH);
#pragma unroll
            for (int q = 0; q < 4; ++q) {
                const float4 a = v00[q], b = v01[q], c = v10[q], d = v11[q];
                acc[q * 4 + 0] += c00 * a.x + c01 * b.x + c10 * c.x + c11 * d.x;
                acc[q * 4 + 1] += c00 * a.y + c01 * b.y + c10 * c.y + c11 * d.y;
                acc[q * 4 + 2] += c00 * a.z + c01 * b.z + c10 * c.z + c11 * d.z;
                acc[q * 4 + 3] += c00 * a.w + c01 * b.w + c10 * c.w + c11 * d.w;
            }
        }
    }

    float4* o = (float4*)(out + (size_t)n * CH);
#pragma unroll
    for (int q = 0; q < 4; ++q) {
        o[q] = make_float4(2.f * acc[q * 4 + 0], 2.f * acc[q * 4 + 1],
                           2.f * acc[q * 4 + 2], 2.f * acc[q * 4 + 3]);
    }
}

// ---------------------- fallback (original layout, no workspace) -----------
__global__ __launch_bounds__(256)
void triplane_direct_kernel(const float* __restrict__ coords,
                            const float* __restrict__ Pu,
                            const float* __restrict__ Pv,
                            const float* __restrict__ Pw,
                            float* __restrict__ out, int N)
{
    const int n = blockIdx.x * blockDim.x + threadIdx.x;
    if (n >= N) return;

    const float cx = coords[3 * n + 0];
    const float cy = coords[3 * n + 1];
    const float cz = coords[3 * n + 2];

    const PlaneGeom Gu = plane_geom(cy, cz, cx);
    const PlaneGeom Gv = plane_geom(cx, cz, cy);
    const PlaneGeom Gw = plane_geom(cx, cy, cz);

    const float*     Ps[3] = { Pu, Pv, Pw };
    const PlaneGeom* Gs[3] = { &Gu, &Gv, &Gw };

    float acc[CH];
#pragma unroll
    for (int ch = 0; ch < CH; ++ch) acc[ch] = 0.f;

    for (int p = 0; p < 3; ++p) {
        const float* P = Ps[p];
        const PlaneGeom& G = *Gs[p];
#pragma unroll
        for (int ch = 0; ch < CH; ++ch) {
            float s = 0.f;
#pragma unroll
            for (int r = 0; r < RR; ++r) {
                const float* base = P + (size_t)(ch * RR + r) * NPIX;
                const float v = G.w00 * base[G.i00] + G.w01 * base[G.i01]
                              + G.w10 * base[G.i10] + G.w11 * base[G.i11];
                s += G.basis[r] * v;
            }
            acc[ch] += s;
        }
    }
#pragma unroll
    for (int ch = 0; ch < CH; ++ch) out[(size_t)n * CH + ch] = 2.f * acc[ch];
}

// ---------------------------------------------------------------------------
extern "C" void kernel_launch(void* const* d_in, const int* in_sizes, int n_in,
                              void* d_out, int out_size, void* d_ws, size_t ws_size,
                              hipStream_t stream)
{
    const float* coords = (const float*)d_in[0];
    const float* Pu     = (const float*)d_in[1];
    const float* Pv     = (const float*)d_in[2];
    const float* Pw     = (const float*)d_in[3];
    float*       out    = (float*)d_out;
    const int    N      = in_sizes[0] / 3;

    const size_t plane_f = (size_t)C144 * NPIX;          // 9,437,184 floats
    const size_t needed  = 3 * plane_f * sizeof(float);  // ~113 MB

    if (ws_size >= needed) {
        float* Tu = (float*)d_ws;
        float* Tv = Tu + plane_f;
        float* Tw = Tv + plane_f;

        const dim3 tgrid(NPIX / TP_PIX);                 // 2048 blocks / plane
        transpose_plane_kernel<<<tgrid, TP_THREADS, 0, stream>>>(Pu, Tu);
        transpose_plane_kernel<<<tgrid, TP_THREADS, 0, stream>>>(Pv, Tv);
        transpose_plane_kernel<<<tgrid, TP_THREADS, 0, stream>>>(Pw, Tw);

        triplane_kernel<<<(N + 255) / 256, 256, 0, stream>>>(coords, Tu, Tv, Tw, out, N);
    } else {
        triplane_direct_kernel<<<(N + 255) / 256, 256, 0, stream>>>(coords, Pu, Pv, Pw, out, N);
    }
}